// HeteroGNN_44109314130257
// MI455X (gfx1250) — compile-verified
//
#include <hip/hip_runtime.h>

typedef __attribute__((ext_vector_type(2))) float v2f;
typedef __attribute__((ext_vector_type(8))) float v8f;

#define D_FEAT 128

// ---------------------------------------------------------------------------
// Degree counting (once; same every layer)
// ---------------------------------------------------------------------------
__global__ void count_edges_kernel(const int* __restrict__ dst, int E,
                                   int* __restrict__ cnt) {
    int e = blockIdx.x * blockDim.x + threadIdx.x;
    if (e < E) atomicAdd(&cnt[dst[e]], 1);
}

__global__ void inv_cnt_kernel(const int* __restrict__ cnt,
                               float* __restrict__ inv, int n) {
    int i = blockIdx.x * blockDim.x + threadIdx.x;
    if (i < n) {
        int c = cnt[i];
        inv[i] = 1.0f / (float)(c > 1 ? c : 1);
    }
}

// ---------------------------------------------------------------------------
// Edge scatter: sum[dst] += x[src].  One wave (32 lanes) per edge; each lane
// handles 4 of the 128 features -> fully coalesced 512B row reads, f32
// hardware atomics into L2 for accumulation.
// ---------------------------------------------------------------------------
__global__ void __launch_bounds__(256)
scatter_sum_kernel(const float* __restrict__ x, const int* __restrict__ src,
                   const int* __restrict__ dst, int E, float* __restrict__ sum) {
    int wave = (int)((blockIdx.x * blockDim.x + threadIdx.x) >> 5);
    int lane = threadIdx.x & 31;
    if (wave >= E) return;
    int s = src[wave];
    int d = dst[wave];
    const float* xs = x + (size_t)s * D_FEAT;
    float* o = sum + (size_t)d * D_FEAT;
#pragma unroll
    for (int i = 0; i < 4; ++i) {
        float v = xs[lane + 32 * i];
        unsafeAtomicAdd(&o[lane + 32 * i], v);
    }
}

// ---------------------------------------------------------------------------
// Fused SAGE layer GEMM:
//   out = leaky( (Asum * inv[row]) @ Wl  +  Xd @ Wr  + bias )
// Block = 16 output rows x 128 cols.  8 waves, wave w owns cols [16w,16w+16).
// Per wave: 32 K-steps of V_WMMA_F32_16X16X4_F32 on two independent
// accumulator chains (ILP), 64 WMMA total.
//
// ISA 7.12.2 layouts (wave32):
//   A 16x4 f32 : lanes 0-15 -> M=lane, K={0,1} in v[0],v[1];
//                lanes 16-31 -> M=lane-16, K={2,3}
//   B 4x16 f32 : lanes 0-15 -> N=lane, K={0,1}; lanes 16-31 -> K={2,3}
//   C/D 16x16  : vgpr j: lanes 0-15 -> (M=j, N=lane);
//                        lanes 16-31 -> (M=j+8, N=lane-16)
// ---------------------------------------------------------------------------
__global__ void __launch_bounds__(256)
sage_gemm_kernel(const float* __restrict__ Asum, const float* __restrict__ inv,
                 const float* __restrict__ Xd, const float* __restrict__ Wl,
                 const float* __restrict__ Wr, const float* __restrict__ bias,
                 float* __restrict__ out, int M) {
    int wave = threadIdx.x >> 5;   // 0..7 -> column tile
    int lane = threadIdx.x & 31;
    int mrow = lane & 15;          // A row within tile / B-C-D column within tile
    int khalf = lane >> 4;         // 0 or 1 -> K offset 0 or 2
    int m0 = blockIdx.x * 16;
    int n0 = wave * 16;

    int m = m0 + mrow;
    bool inb = (m < M);
    int msafe = inb ? m : 0;
    float sc = inb ? inv[msafe] : 0.0f;
    const float* arow1 = Asum + (size_t)msafe * D_FEAT;  // summed messages
    const float* arow2 = Xd + (size_t)msafe * D_FEAT;    // dst features
    float zmask = inb ? 1.0f : 0.0f;

    v8f acc1 = {};
    v8f acc2 = {};

#pragma unroll
    for (int kk = 0; kk < D_FEAT; kk += 4) {
        int k = kk + 2 * khalf;
        v2f a1, a2, b1, b2;
        a1.x = arow1[k] * sc;
        a1.y = arow1[k + 1] * sc;
        a2.x = arow2[k] * zmask;
        a2.y = arow2[k + 1] * zmask;
        int n = n0 + mrow;
        b1.x = Wl[k * D_FEAT + n];
        b1.y = Wl[(k + 1) * D_FEAT + n];
        b2.x = Wr[k * D_FEAT + n];
        b2.y = Wr[(k + 1) * D_FEAT + n];
        acc1 = __builtin_amdgcn_wmma_f32_16x16x4_f32(
            false, a1, false, b1, (short)0, acc1, false, false);
        acc2 = __builtin_amdgcn_wmma_f32_16x16x4_f32(
            false, a2, false, b2, (short)0, acc2, false, false);
    }

    float bn = bias[n0 + mrow];
#pragma unroll
    for (int j = 0; j < 8; ++j) {
        int mo = m0 + j + khalf * 8;
        if (mo < M) {
            float v = acc1[j] + acc2[j] + bn;
            v = (v >= 0.0f) ? v : 0.1f * v;
            out[(size_t)mo * D_FEAT + n0 + mrow] = v;
        }
    }
}

// ---------------------------------------------------------------------------
// Final projection: out[n] = dot(xc[n,:], W_lin) + b_lin.  Wave per node.
// ---------------------------------------------------------------------------
__global__ void __launch_bounds__(256)
final_proj_kernel(const float* __restrict__ xc, const float* __restrict__ Wlin,
                  const float* __restrict__ blin, float* __restrict__ out, int M) {
    int node = (int)((blockIdx.x * blockDim.x + threadIdx.x) >> 5);
    int lane = threadIdx.x & 31;
    if (node >= M) return;
    const float* xr = xc + (size_t)node * D_FEAT;
    float s = 0.0f;
#pragma unroll
    for (int i = 0; i < 4; ++i) s += xr[lane + 32 * i] * Wlin[lane + 32 * i];
#pragma unroll
    for (int off = 16; off > 0; off >>= 1) s += __shfl_xor(s, off, 32);
    if (lane == 0) out[node] = s + blin[0];
}

// ---------------------------------------------------------------------------
// Host orchestration
// ---------------------------------------------------------------------------
extern "C" void kernel_launch(void* const* d_in, const int* in_sizes, int n_in,
                              void* d_out, int out_size, void* d_ws, size_t ws_size,
                              hipStream_t stream) {
    const float* x_clients = (const float*)d_in[0];
    const float* x_agg     = (const float*)d_in[1];
    const int* c2a_src = (const int*)d_in[2];
    const int* c2a_dst = (const int*)d_in[3];
    const int* a2c_src = (const int*)d_in[4];
    const int* a2c_dst = (const int*)d_in[5];
    const float* Wl_c2a = (const float*)d_in[6];
    const float* Wr_c2a = (const float*)d_in[7];
    const float* b_c2a  = (const float*)d_in[8];
    const float* Wl_a2c = (const float*)d_in[9];
    const float* Wr_a2c = (const float*)d_in[10];
    const float* b_a2c  = (const float*)d_in[11];
    const float* W_lin  = (const float*)d_in[12];
    const float* b_lin  = (const float*)d_in[13];

    const int NC = in_sizes[0] / D_FEAT;       // 100000
    const int NA = in_sizes[1] / D_FEAT;       // 1000
    const int E  = in_sizes[2];                // 1600000
    const int L  = in_sizes[8] / D_FEAT;       // 3

    // workspace carve-up (floats)
    char* ws = (char*)d_ws;
    size_t off = 0;
    auto take = [&](size_t bytes) {
        char* p = ws + off;
        off += (bytes + 255) & ~(size_t)255;
        return p;
    };
    float* xcbuf[2] = {(float*)take((size_t)NC * D_FEAT * 4),
                       (float*)take((size_t)NC * D_FEAT * 4)};
    float* xabuf[2] = {(float*)take((size_t)NA * D_FEAT * 4),
                       (float*)take((size_t)NA * D_FEAT * 4)};
    float* sum_c = (float*)take((size_t)NC * D_FEAT * 4);
    float* sum_a = (float*)take((size_t)NA * D_FEAT * 4);
    float* inv_c = (float*)take((size_t)NC * 4);
    float* inv_a = (float*)take((size_t)NA * 4);
    int* icnt_c = (int*)take((size_t)NC * 4);
    int* icnt_a = (int*)take((size_t)NA * 4);

    // degree counts (constant across layers)
    hipMemsetAsync(icnt_c, 0, (size_t)NC * 4, stream);
    hipMemsetAsync(icnt_a, 0, (size_t)NA * 4, stream);
    {
        dim3 b(256), g((E + 255) / 256);
        count_edges_kernel<<<g, b, 0, stream>>>(c2a_dst, E, icnt_a);
        count_edges_kernel<<<g, b, 0, stream>>>(a2c_dst, E, icnt_c);
        inv_cnt_kernel<<<dim3((NA + 255) / 256), b, 0, stream>>>(icnt_a, inv_a, NA);
        inv_cnt_kernel<<<dim3((NC + 255) / 256), b, 0, stream>>>(icnt_c, inv_c, NC);
    }

    const float* cur_xc = x_clients;
    const float* cur_xa = x_agg;

    for (int l = 0; l < L; ++l) {
        float* next_xc = xcbuf[l & 1];
        float* next_xa = xabuf[l & 1];

        hipMemsetAsync(sum_a, 0, (size_t)NA * D_FEAT * 4, stream);
        hipMemsetAsync(sum_c, 0, (size_t)NC * D_FEAT * 4, stream);

        // gather/scatter (wave per edge, 8 waves per block)
        dim3 sb(256), sg((E + 7) / 8);
        scatter_sum_kernel<<<sg, sb, 0, stream>>>(cur_xc, c2a_src, c2a_dst, E, sum_a);
        scatter_sum_kernel<<<sg, sb, 0, stream>>>(cur_xa, a2c_src, a2c_dst, E, sum_c);

        // fused mean-scale + dual GEMM + bias + leaky
        sage_gemm_kernel<<<dim3((NA + 15) / 16), dim3(256), 0, stream>>>(
            sum_a, inv_a, cur_xa,
            Wl_c2a + (size_t)l * D_FEAT * D_FEAT,
            Wr_c2a + (size_t)l * D_FEAT * D_FEAT,
            b_c2a + (size_t)l * D_FEAT, next_xa, NA);
        sage_gemm_kernel<<<dim3((NC + 15) / 16), dim3(256), 0, stream>>>(
            sum_c, inv_c, cur_xc,
            Wl_a2c + (size_t)l * D_FEAT * D_FEAT,
            Wr_a2c + (size_t)l * D_FEAT * D_FEAT,
            b_a2c + (size_t)l * D_FEAT, next_xc, NC);

        cur_xc = next_xc;
        cur_xa = next_xa;
    }

    final_proj_kernel<<<dim3((NC + 7) / 8), dim3(256), 0, stream>>>(
        cur_xc, W_lin, b_lin, (float*)d_out, NC);
}